// BahdanauAttention_43533788513024
// MI455X (gfx1250) — compile-verified
//
#include <hip/hip_runtime.h>
#include <hip/hip_bf16.h>

// Shapes from the reference
#define B_  64
#define T_  2048
#define QD_ 1024
#define AD_ 256

typedef float v2f __attribute__((ext_vector_type(2)));
typedef float v4f __attribute__((ext_vector_type(4)));
typedef float v8f __attribute__((ext_vector_type(8)));

// ---------------------------------------------------------------------------
// Kernel 1: processed_query[b][a] = sum_q query[b][q] * Wq[a][q]
// f32 WMMA 16x16x4. One wave per 16x16 output tile.
// Grid: 64 blocks (4 M-tiles x 16 N-tiles) x 32 threads.
//
// A-matrix (16x4 f32) layout per ISA: lane l holds row m = l&15;
//   VGPR0 = K = 2*(l>>4), VGPR1 = K = 2*(l>>4)+1  -> contiguous float2 in q.
// B-matrix (4x16 f32): lane l holds col n = l&15 with the same K pairing.
//   B[k][n] = Wq[n_global][k_global] (contraction over q), also float2.
// C/D (16x16 f32): lane l, VGPR j -> D[8*(l>>4)+j][l&15].
// ---------------------------------------------------------------------------
__global__ void pq_gemm_wmma(const float* __restrict__ query,
                             const float* __restrict__ Wq,
                             float* __restrict__ pq) {
    const int lane  = threadIdx.x;          // 0..31 (wave32)
    const int mtile = blockIdx.x >> 4;      // 0..3
    const int ntile = blockIdx.x & 15;      // 0..15
    const int mrow  = mtile * 16 + (lane & 15);   // batch row for A
    const int ncol  = ntile * 16 + (lane & 15);   // 'a' column for B
    const int kh    = (lane >> 4) * 2;            // k sub-offset 0 or 2

    const float* qrow = query + (size_t)mrow * QD_ + kh;
    const float* wrow = Wq    + (size_t)ncol * QD_ + kh;

    v8f c = {0.f, 0.f, 0.f, 0.f, 0.f, 0.f, 0.f, 0.f};
#pragma unroll 4
    for (int k0 = 0; k0 < QD_; k0 += 4) {
        v2f a = *(const v2f*)(qrow + k0);
        v2f b = *(const v2f*)(wrow + k0);
        c = __builtin_amdgcn_wmma_f32_16x16x4_f32(
                /*neg_a=*/false, a, /*neg_b=*/false, b,
                /*c_mod=*/(short)0, c, /*reuse_a=*/false, /*reuse_b=*/false);
    }

    const int mBase = mtile * 16 + 8 * (lane >> 4);
    const int nOut  = ntile * 16 + (lane & 15);
#pragma unroll
    for (int j = 0; j < 8; ++j)
        pq[(size_t)(mBase + j) * AD_ + nOut] = c[j];
}

// ---------------------------------------------------------------------------
// Kernel 2 (hot, HBM-bound): energies[b][t] =
//     mask * sum_a tanh(pq[b][a] + enc[b][t][a]) * v[a]
// One wave processes one t-row (256 f32 = 1 KB) per iteration:
//   lane l loads enc[row + 4l .. 4l+3] and enc[row + 128+4l .. 128+4l+3]
//   -> two fully coalesced b128 loads covering the whole row.
// pq/v slices are loop-invariant per lane -> kept in registers.
// Grid: B_*16 = 1024 blocks x 256 threads (8 waves); each wave does 16 rows.
// ---------------------------------------------------------------------------
__global__ void energies_kernel(const float* __restrict__ enc,
                                const int*   __restrict__ chars,
                                const float* __restrict__ pq,
                                const float* __restrict__ v,
                                float* __restrict__ energies) {
    const int b     = blockIdx.x >> 4;      // 16 chunks per batch row
    const int chunk = blockIdx.x & 15;
    const int lane  = threadIdx.x & 31;
    const int wave  = threadIdx.x >> 5;     // 0..7
    const int a0    = lane * 4;             // 0..124

    const v4f v0 = *(const v4f*)(v + a0);
    const v4f v1 = *(const v4f*)(v + 128 + a0);
    const v4f p0 = *(const v4f*)(pq + (size_t)b * AD_ + a0);
    const v4f p1 = *(const v4f*)(pq + (size_t)b * AD_ + 128 + a0);

    const int tBase = chunk * (T_ / 16);    // 128 rows per block

    for (int ti = wave; ti < T_ / 16; ti += 8) {
        const int t = tBase + ti;
        const float* row = enc + ((size_t)b * T_ + t) * AD_;

        // Prefetch this wave's next row (8 rows ahead); speculative, safe.
        __builtin_prefetch(row + 8 * AD_, 0, 0);

        // Read-once stream: non-temporal to avoid polluting L2.
        v4f e0 = __builtin_nontemporal_load((const v4f*)(row + a0));
        v4f e1 = __builtin_nontemporal_load((const v4f*)(row + 128 + a0));

        float acc;
        acc = __builtin_amdgcn_tanhf(p0.x + e0.x) * v0.x;
        acc = fmaf(__builtin_amdgcn_tanhf(p0.y + e0.y), v0.y, acc);
        acc = fmaf(__builtin_amdgcn_tanhf(p0.z + e0.z), v0.z, acc);
        acc = fmaf(__builtin_amdgcn_tanhf(p0.w + e0.w), v0.w, acc);
        acc = fmaf(__builtin_amdgcn_tanhf(p1.x + e1.x), v1.x, acc);
        acc = fmaf(__builtin_amdgcn_tanhf(p1.y + e1.y), v1.y, acc);
        acc = fmaf(__builtin_amdgcn_tanhf(p1.z + e1.z), v1.z, acc);
        acc = fmaf(__builtin_amdgcn_tanhf(p1.w + e1.w), v1.w, acc);

        // wave32 xor-shuffle reduction (5 steps)
#pragma unroll
        for (int off = 16; off > 0; off >>= 1)
            acc += __shfl_xor(acc, off, 32);

        if (lane == 0) {
            const float m = (chars[(size_t)b * T_ + t] != 0) ? 1.0f : 0.0f;
            energies[(size_t)b * T_ + t] = acc * m;
        }
    }
}

// ---------------------------------------------------------------------------
// Kernel 3: row softmax over T (in place on d_out). One block per batch row.
// 256 threads, 8 elements each, LDS tree reductions for max and sum.
// ---------------------------------------------------------------------------
__global__ void softmax_kernel(float* buf) {
    __shared__ float red[256];
    const int b   = blockIdx.x;
    const int tid = threadIdx.x;
    float* rowp = buf + (size_t)b * T_;

    float vals[8];
    float mx = -1e30f;
#pragma unroll
    for (int i = 0; i < 8; ++i) {
        vals[i] = rowp[tid + i * 256];
        mx = fmaxf(mx, vals[i]);
    }
    red[tid] = mx;
    __syncthreads();
    for (int s = 128; s > 0; s >>= 1) {
        if (tid < s) red[tid] = fmaxf(red[tid], red[tid + s]);
        __syncthreads();
    }
    mx = red[0];
    __syncthreads();

    float sum = 0.f;
#pragma unroll
    for (int i = 0; i < 8; ++i) {
        vals[i] = __expf(vals[i] - mx);
        sum += vals[i];
    }
    red[tid] = sum;
    __syncthreads();
    for (int s = 128; s > 0; s >>= 1) {
        if (tid < s) red[tid] += red[tid + s];
        __syncthreads();
    }
    const float inv = 1.0f / red[0];
#pragma unroll
    for (int i = 0; i < 8; ++i)
        rowp[tid + i * 256] = vals[i] * inv;
}

// ---------------------------------------------------------------------------
// Inputs (setup_inputs order):
//   0: encoder_seq_proj (B,T,AD) f32
//   1: query            (B,QD)   f32
//   2: chars            (B,T)    int
//   3: t                scalar   (unused)
//   4: Wq               (AD,QD)  f32
//   5: v                (AD,)    f32
// Output: alignment (B,T) f32
// Workspace: pq (B*AD f32 = 64 KB)
// ---------------------------------------------------------------------------
extern "C" void kernel_launch(void* const* d_in, const int* in_sizes, int n_in,
                              void* d_out, int out_size, void* d_ws, size_t ws_size,
                              hipStream_t stream) {
    const float* enc   = (const float*)d_in[0];
    const float* query = (const float*)d_in[1];
    const int*   chars = (const int*)d_in[2];
    const float* Wq    = (const float*)d_in[4];
    const float* v     = (const float*)d_in[5];
    float* out = (float*)d_out;
    float* pq  = (float*)d_ws;   // B_*AD_ floats

    pq_gemm_wmma<<<dim3(64), dim3(32), 0, stream>>>(query, Wq, pq);
    energies_kernel<<<dim3(B_ * 16), dim3(256), 0, stream>>>(enc, chars, pq, v, out);
    softmax_kernel<<<dim3(B_), dim3(256), 0, stream>>>(out);
}